// BaseGIN_54752243090034
// MI455X (gfx1250) — compile-verified
//
#include <hip/hip_runtime.h>

// GIN forward, 3 layers, fused for MI455X (gfx1250, wave32).
// Strategy: whole working set (~100MB) is L2-resident (192MB L2).
// Per layer: k1 init hpre=(1+eps)*x + zero BN stats
//            k2 edge scatter: hpre[dst] += x[src]*w   (f32 global atomics)
//            k3 WMMA MLP: h2 = relu(hpre@W1+b1)@W2+b2, fused BN sum/sumsq
//            k4 BN + ReLU + residual
// GEMMs use V_WMMA_F32_16X16X4_F32 (full f32 precision, K=4 x 32 steps).

#define NNODES   50000
#define NEDGES   800000
#define DIM      128
#define NTILES_M 3125        // NNODES / 16, exact
#define BN_EPS_F 1e-5f

typedef float v2f __attribute__((ext_vector_type(2)));
typedef float v8f __attribute__((ext_vector_type(8)));

// ---------------- k1: hpre = (1+eps)*x ; zero BN stats ----------------
__global__ __launch_bounds__(256)
void gin_init_kernel(const float* __restrict__ xin, const float* __restrict__ eps,
                     int layer, float* __restrict__ hpre, float* __restrict__ stats)
{
    const int gid = blockIdx.x * blockDim.x + threadIdx.x;   // one float4 each
    if (gid < 2 * DIM) stats[gid] = 0.0f;                    // sums + sumsq
    const float s = 1.0f + eps[layer];
    float4 v = ((const float4*)xin)[gid];
    v.x *= s; v.y *= s; v.z *= s; v.w *= s;
    ((float4*)hpre)[gid] = v;
}

// ---------------- k2: scatter-add messages ----------------
__global__ __launch_bounds__(256)
void gin_scatter_kernel(const float* __restrict__ x, const long long* __restrict__ ei,
                        const float* __restrict__ ew, float* __restrict__ hpre)
{
    const int e    = (int)((blockIdx.x * blockDim.x + threadIdx.x) >> 5); // wave per edge
    const int lane = threadIdx.x & 31;
    if (e >= NEDGES) return;
    const int   src = (int)ei[e];            // edge_index[0][e] (int64)
    const int   dst = (int)ei[NEDGES + e];   // edge_index[1][e]
    const float w   = ew[e];
    const float* xs = x    + (size_t)src * DIM;
    float*       hd = hpre + (size_t)dst * DIM;
#pragma unroll
    for (int i = 0; i < 4; ++i) {
        const int d = lane + 32 * i;
        unsafeAtomicAdd(&hd[d], xs[d] * w);  // -> global_atomic_add_f32
    }
}

// ---------------- k3: fused 2-GEMM MLP via f32 WMMA + BN stat reduce ----------------
// One wave per 16-row node tile. A-layout (f32 16x4): lane = m + 16*(k>=2), vgpr = k&1.
// C-layout (f32 16x16): lane = n + 16*(m>=8), vgpr = m&7.
__global__ __launch_bounds__(128)
void gin_mlp_kernel(const float* __restrict__ hpre,
                    const float* __restrict__ W1, const float* __restrict__ b1,
                    const float* __restrict__ W2, const float* __restrict__ b2,
                    float* __restrict__ h2, float* __restrict__ stats)
{
    __shared__ __align__(16) float lds[4][16 * DIM];   // 8KB per wave, 32KB total
    const int wave = threadIdx.x >> 5;
    const int lane = threadIdx.x & 31;
    const int tile = blockIdx.x * 4 + wave;
    const bool active = (tile < NTILES_M);             // wave-uniform: EXEC all-ones inside
    const int n  = lane & 15;                          // col in 16-tile / row for A loads
    const int kh = lane >> 4;                          // K-half (A/B) / M-half (C/D)
    float* my = lds[wave];
    v8f acc[8];

    if (active) {
        const float* arow = hpre + (size_t)(tile * 16 + n) * DIM;
        // ---- GEMM1: H1 = Hpre(16x128) @ W1(128x128) + b1 ----
#pragma unroll
        for (int nt = 0; nt < 8; ++nt) {
            const float bias = b1[nt * 16 + n];
#pragma unroll
            for (int v = 0; v < 8; ++v) acc[nt][v] = bias;
        }
        for (int kk = 0; kk < 32; ++kk) {              // K steps of 4
            const v2f a = *(const v2f*)(arow + 4 * kk + 2 * kh);       // global_load_b64
            const float* wcol = W1 + (size_t)(4 * kk + 2 * kh) * DIM + n;
#pragma unroll
            for (int nt = 0; nt < 8; ++nt) {           // 8 independent WMMA chains
                v2f b;
                b.x = wcol[nt * 16];
                b.y = wcol[nt * 16 + DIM];
                acc[nt] = __builtin_amdgcn_wmma_f32_16x16x4_f32(
                    false, a, false, b, (short)0, acc[nt], false, false);
            }
        }
        // ReLU, stage 16x128 H1 tile row-major into this wave's LDS slice
#pragma unroll
        for (int nt = 0; nt < 8; ++nt) {
#pragma unroll
            for (int v = 0; v < 8; ++v) {
                float val = acc[nt][v];
                val = val > 0.0f ? val : 0.0f;
                my[(v + 8 * kh) * DIM + nt * 16 + n] = val;
            }
        }
    }
    __syncthreads();    // order LDS write -> cross-lane LDS read
    if (active) {
        const int row0 = tile * 16;
        // ---- GEMM2: H2 = H1(16x128) @ W2(128x128) + b2 ----
#pragma unroll
        for (int nt = 0; nt < 8; ++nt) {
            const float bias = b2[nt * 16 + n];
#pragma unroll
            for (int v = 0; v < 8; ++v) acc[nt][v] = bias;
        }
        const float* arow2 = my + n * DIM;
        for (int kk = 0; kk < 32; ++kk) {
            const v2f a = *(const v2f*)(arow2 + 4 * kk + 2 * kh);      // ds_load_b64
            const float* wcol = W2 + (size_t)(4 * kk + 2 * kh) * DIM + n;
#pragma unroll
            for (int nt = 0; nt < 8; ++nt) {
                v2f b;
                b.x = wcol[nt * 16];
                b.y = wcol[nt * 16 + DIM];
                acc[nt] = __builtin_amdgcn_wmma_f32_16x16x4_f32(
                    false, a, false, b, (short)0, acc[nt], false, false);
            }
        }
        // Store H2 + in-register BN partial sums (8 rows per lane, per column)
#pragma unroll
        for (int nt = 0; nt < 8; ++nt) {
            float psum = 0.0f, psq = 0.0f;
#pragma unroll
            for (int v = 0; v < 8; ++v) {
                const float val = acc[nt][v];
                h2[(size_t)(row0 + v + 8 * kh) * DIM + nt * 16 + n] = val;
                psum += val;
                psq  += val * val;
            }
            unsafeAtomicAdd(&stats[nt * 16 + n],       psum);
            unsafeAtomicAdd(&stats[DIM + nt * 16 + n], psq);
        }
    }
}

// ---------------- k4: BatchNorm + ReLU + residual ----------------
__global__ __launch_bounds__(256)
void gin_bn_kernel(const float* __restrict__ xin, const float* __restrict__ h2,
                   const float* __restrict__ stats, const float* __restrict__ gamma,
                   const float* __restrict__ beta, float* __restrict__ xout)
{
    const int gid  = blockIdx.x * blockDim.x + threadIdx.x;  // one float4 each
    const int base = gid * 4;
    const int d    = base & (DIM - 1);
    const float invN = 1.0f / (float)NNODES;
    const float4 h  = ((const float4*)h2)[gid];
    const float4 xv = ((const float4*)xin)[gid];
    const float hv[4]  = {h.x, h.y, h.z, h.w};
    const float xvv[4] = {xv.x, xv.y, xv.z, xv.w};
    float o[4];
#pragma unroll
    for (int j = 0; j < 4; ++j) {
        const float mu  = stats[d + j] * invN;
        const float var = stats[DIM + d + j] * invN - mu * mu;
        float val = gamma[d + j] * (hv[j] - mu) * rsqrtf(var + BN_EPS_F) + beta[d + j];
        val = val > 0.0f ? val : 0.0f;
        o[j] = xvv[j] + val;
    }
    float4 r; r.x = o[0]; r.y = o[1]; r.z = o[2]; r.w = o[3];
    ((float4*)xout)[gid] = r;
}

extern "C" void kernel_launch(void* const* d_in, const int* in_sizes, int n_in,
                              void* d_out, int out_size, void* d_ws, size_t ws_size,
                              hipStream_t stream)
{
    (void)in_sizes; (void)n_in; (void)out_size; (void)ws_size;
    const float*     x     = (const float*)d_in[0];
    const long long* ei    = (const long long*)d_in[1];   // int64 (2, NEDGES)
    const float*     ew    = (const float*)d_in[3];
    const float*     W1    = (const float*)d_in[4];
    const float*     b1    = (const float*)d_in[5];
    const float*     W2    = (const float*)d_in[6];
    const float*     b2    = (const float*)d_in[7];
    const float*     eps   = (const float*)d_in[8];
    const float*     gamma = (const float*)d_in[9];
    const float*     beta  = (const float*)d_in[10];
    float* out = (float*)d_out;

    const size_t ND = (size_t)NNODES * DIM;               // 6.4M floats
    float* hpre  = (float*)d_ws;                          // [ND]
    float* h2    = hpre + ND;                             // [ND]
    float* xbuf  = h2 + ND;                               // [ND]
    float* stats = xbuf + ND;                             // [2*DIM]

    const int elem4Blocks   = (int)(ND / 4 / 256);        // 6250 exact
    const int scatterBlocks = NEDGES / 8;                 // 100000 (8 waves/block)
    const int mlpBlocks     = (NTILES_M + 3) / 4;         // 782 (4 waves/block)

    for (int layer = 0; layer < 3; ++layer) {
        const float* xin = (layer == 0) ? x : xbuf;
        float*       xo  = (layer == 2) ? out : xbuf;     // in-place residual OK (elementwise)
        gin_init_kernel<<<elem4Blocks, 256, 0, stream>>>(xin, eps, layer, hpre, stats);
        gin_scatter_kernel<<<scatterBlocks, 256, 0, stream>>>(xin, ei, ew, hpre);
        gin_mlp_kernel<<<mlpBlocks, 128, 0, stream>>>(
            hpre,
            W1 + (size_t)layer * DIM * DIM, b1 + (size_t)layer * DIM,
            W2 + (size_t)layer * DIM * DIM, b2 + (size_t)layer * DIM,
            h2, stats);
        gin_bn_kernel<<<elem4Blocks, 256, 0, stream>>>(
            xin, h2, stats, gamma + (size_t)layer * DIM, beta + (size_t)layer * DIM, xo);
    }
}